// AmbientOrthogonalGroupGenerative_72456098283814
// MI455X (gfx1250) — compile-verified
//
#include <hip/hip_runtime.h>
#include <cstdint>

// ---------------------------------------------------------------------------
// Polar-factor (nearest orthogonal) + tangent projection for 2M 3x3 matrices.
// vt = 0.5*(v - Q v^T Q), Q = polar(x) via scaled Newton iteration.
// HBM-streaming kernel: double-buffered global->LDS staging with CDNA5
// b128 async-tensor copies (ASYNCcnt), 4 matrices (144 B, 16B-aligned) per
// thread so both async loads and output stores run at b128 width.
// ---------------------------------------------------------------------------

#define TPB 256
#define THR_BYTES 144                      // 4 matrices * 36 B per thread per array
#define ARR_BYTES (TPB * THR_BYTES)        // 36864 B per array per tile
#define BUF_BYTES (2 * ARR_BYTES)          // 73728 B (x-segment + v-segment)
#define ARR_FLOATS (ARR_BYTES / 4)         // 9216
#define BUF_FLOATS (BUF_BYTES / 4)         // 18432
#define LDS_FLOATS (2 * BUF_FLOATS)        // 36864 floats = 147456 B

// One 16-byte async copy global->LDS. INST_OFFSET advances BOTH the LDS
// address (VDST vgpr) and the global address (saddr+vaddr) per the ISA
// pseudocode, so one LDS-addr VGPR + one offset VGPR serve the whole 144 B.
#define ASYNC_LD_B128(ldsaddr, voff, sbase, IMM)                              \
  asm volatile("global_load_async_to_lds_b128 %0, %1, %2 offset:" IMM         \
               :: "v"(ldsaddr), "v"(voff), "s"(sbase) : "memory")

#define ASYNC_LD_144B(ldsaddr, voff, sbase)                                   \
  do {                                                                        \
    ASYNC_LD_B128((ldsaddr), (voff), (sbase), "0");                           \
    ASYNC_LD_B128((ldsaddr), (voff), (sbase), "16");                          \
    ASYNC_LD_B128((ldsaddr), (voff), (sbase), "32");                          \
    ASYNC_LD_B128((ldsaddr), (voff), (sbase), "48");                          \
    ASYNC_LD_B128((ldsaddr), (voff), (sbase), "64");                          \
    ASYNC_LD_B128((ldsaddr), (voff), (sbase), "80");                          \
    ASYNC_LD_B128((ldsaddr), (voff), (sbase), "96");                          \
    ASYNC_LD_B128((ldsaddr), (voff), (sbase), "112");                         \
    ASYNC_LD_B128((ldsaddr), (voff), (sbase), "128");                         \
  } while (0)

#define WAIT_ASYNC(IMM) asm volatile("s_wait_asynccnt " IMM ::: "memory")

__device__ __forceinline__ float fast_rcp(float x) { return __builtin_amdgcn_rcpf(x); }
__device__ __forceinline__ float fast_rsq(float x) { return __builtin_amdgcn_rsqf(x); }

// Scaled Newton for the polar factor of a 3x3 (in place):
// X <- 0.5*(g*X + (1/g)*X^{-T}), g = (||X^{-T}||_F/||X||_F)^{1/2},
// X^{-T} = cofactor(X)/det(X). Scaling first 4 of 6 iterations; quadratic
// convergence makes this ample for fp32.
__device__ __forceinline__ void polar3x3(float x[9]) {
#pragma unroll
  for (int it = 0; it < 6; ++it) {
    const float c00 = x[4] * x[8] - x[5] * x[7];
    const float c01 = x[5] * x[6] - x[3] * x[8];
    const float c02 = x[3] * x[7] - x[4] * x[6];
    const float c10 = x[2] * x[7] - x[1] * x[8];
    const float c11 = x[0] * x[8] - x[2] * x[6];
    const float c12 = x[1] * x[6] - x[0] * x[7];
    const float c20 = x[1] * x[5] - x[2] * x[4];
    const float c21 = x[2] * x[3] - x[0] * x[5];
    const float c22 = x[0] * x[4] - x[1] * x[3];
    const float det = x[0] * c00 + x[1] * c01 + x[2] * c02;
    const float rdet = fast_rcp(det);
    float y[9] = {c00 * rdet, c01 * rdet, c02 * rdet,
                  c10 * rdet, c11 * rdet, c12 * rdet,
                  c20 * rdet, c21 * rdet, c22 * rdet};
    float g = 1.0f, gi = 1.0f;
    if (it < 4) {
      float fx = 0.0f, fy = 0.0f;
#pragma unroll
      for (int k = 0; k < 9; ++k) {
        fx = fmaf(x[k], x[k], fx);
        fy = fmaf(y[k], y[k], fy);
      }
      const float q = fy * fast_rcp(fx);   // (||Y||/||X||)^2
      g  = fast_rsq(fast_rsq(q));          // q^(1/4)
      gi = fast_rcp(g);
    }
#pragma unroll
    for (int k = 0; k < 9; ++k) x[k] = 0.5f * (g * x[k] + gi * y[k]);
  }
}

__global__ __launch_bounds__(TPB) void
AmbientOrthogonalGroupGenerative_72456098283814_kernel(
    const float* __restrict__ xg, const float* __restrict__ vg,
    float* __restrict__ outg, int n, int numTiles) {
  __shared__ __align__(16) float smem[LDS_FLOATS];  // [buf][x|v][thread][36]

  const int tid = threadIdx.x;
  const unsigned ldsBase = (unsigned)(uintptr_t)(&smem[0]);
  const unsigned ldsThr = ldsBase + (unsigned)tid * THR_BYTES;  // x slot, buf 0

  const uint64_t xb = (uint64_t)(uintptr_t)xg;
  const uint64_t vb = (uint64_t)(uintptr_t)vg;
  const unsigned maxOff = (unsigned)(n * 36 - THR_BYTES);  // keep 144B window in-bounds

  int tile = (int)blockIdx.x;
  if (tile >= numTiles) return;

  // Prologue: async-stage this block's first tile into buffer 0.
  {
    unsigned boff = (unsigned)(tile * TPB + tid) * (unsigned)THR_BYTES;
    if (boff > maxOff) boff = maxOff;  // clamp: EXEC stays full, ASYNCcnt exact
    ASYNC_LD_144B(ldsThr, boff, xb);
    ASYNC_LD_144B(ldsThr + ARR_BYTES, boff, vb);
  }

  int buf = 0;
  for (; tile < numTiles; tile += (int)gridDim.x) {
    const int nextTile = tile + (int)gridDim.x;
    if (nextTile < numTiles) {
      // Prefetch next tile, then wait until only those 18 ops remain:
      // async loads complete in order => current tile's 18 ops are done.
      unsigned nboff = (unsigned)(nextTile * TPB + tid) * (unsigned)THR_BYTES;
      if (nboff > maxOff) nboff = maxOff;
      const unsigned nlds = ldsThr + (unsigned)((buf ^ 1) * BUF_BYTES);
      ASYNC_LD_144B(nlds, nboff, xb);
      ASYNC_LD_144B(nlds + ARR_BYTES, nboff, vb);
      WAIT_ASYNC("0x12");  // ASYNCcnt <= 18
    } else {
      WAIT_ASYNC("0x0");   // drain
    }

    // ---- compute 4 matrices from LDS (own slots; no barrier needed) ----
    const float* sx = &smem[buf * BUF_FLOATS + tid * 36];
    const float* sv = sx + ARR_FLOATS;

    union Out36 { float4 q[9]; float f[36]; } O;

#pragma unroll
    for (int m = 0; m < 4; ++m) {
      float X[9], V[9];
#pragma unroll
      for (int k = 0; k < 9; ++k) { X[k] = sx[m * 9 + k]; V[k] = sv[m * 9 + k]; }

      polar3x3(X);  // X is now Q = polar factor (U V^T)

      // A = Q * V^T : A[i][k] = sum_j Q[i][j] * V[k][j]
      float A[9];
#pragma unroll
      for (int i = 0; i < 3; ++i)
#pragma unroll
        for (int k = 0; k < 3; ++k)
          A[3 * i + k] = fmaf(X[3 * i + 0], V[3 * k + 0],
                         fmaf(X[3 * i + 1], V[3 * k + 1],
                              X[3 * i + 2] * V[3 * k + 2]));

      // out = 0.5 * (V - A * Q)
#pragma unroll
      for (int i = 0; i < 3; ++i)
#pragma unroll
        for (int l = 0; l < 3; ++l) {
          const float b = fmaf(A[3 * i + 0], X[0 + l],
                          fmaf(A[3 * i + 1], X[3 + l],
                               A[3 * i + 2] * X[6 + l]));
          O.f[m * 9 + 3 * i + l] = 0.5f * (V[3 * i + l] - b);
        }
    }

    // ---- store: 144 B per thread, 16B-aligned => 9 x b128 ----
    const int q = tile * TPB + tid;     // quad index
    const int mbase = q * 4;            // first matrix of this thread
    if (mbase + 3 < n) {
      float4* oq = (float4*)(outg + (size_t)mbase * 9);
#pragma unroll
      for (int k = 0; k < 9; ++k) oq[k] = O.q[k];
    } else if (mbase < n) {
      // rare partial tail quad: scalar stores for the valid matrices
      float* o = outg + (size_t)mbase * 9;
      const int nv = (n - mbase) * 9;
      for (int k = 0; k < nv; ++k) o[k] = O.f[k];
    }

    buf ^= 1;
  }
}

extern "C" void kernel_launch(void* const* d_in, const int* in_sizes, int n_in,
                              void* d_out, int out_size, void* d_ws, size_t ws_size,
                              hipStream_t stream) {
  (void)n_in; (void)out_size; (void)d_ws; (void)ws_size;
  const float* x = (const float*)d_in[0];
  const float* v = (const float*)d_in[1];
  float* out = (float*)d_out;

  const int n = in_sizes[0] / 9;                  // number of 3x3 matrices
  const int nQuads = (n + 3) / 4;                 // 4 matrices per thread
  const int numTiles = (nQuads + TPB - 1) / TPB;
  int blocks = numTiles < 1024 ? numTiles : 1024; // >=2 tiles/block => pipelined

  AmbientOrthogonalGroupGenerative_72456098283814_kernel
      <<<dim3(blocks), dim3(TPB), 0, stream>>>(x, v, out, n, numTiles);
}